// MOE_91139206021768
// MI455X (gfx1250) — compile-verified
//
#include <hip/hip_runtime.h>
#include <hip/hip_bf16.h>

// ---------------------------------------------------------------------------
// MoE forward for MI455X (gfx1250, wave32).
//   x  [N=8192, D=1024] fp32
//   Wr [E=8, D]         fp32 router weights
//   br [E]              fp32
//   We [E, F=1024, D]   fp32 expert weights
//   be [E, F]           fp32
//   capacity = 1024
// Memory-bound (~100 MB traffic vs 17.2 GFLOP) -> keep fp32 and use
// V_WMMA_F32_16X16X4_F32 for the expert GEMMs.  B operand (We tiles) is
// staged into LDS by the Tensor Data Mover (tensor_load_to_lds, TENSORcnt),
// double-buffered so the DMA of chunk k+1 overlaps WMMAs on chunk k.
// ---------------------------------------------------------------------------

typedef __attribute__((ext_vector_type(2))) float v2f;
typedef __attribute__((ext_vector_type(8))) float v8f;
typedef __attribute__((ext_vector_type(4))) unsigned int v4u;
typedef __attribute__((ext_vector_type(8))) int v8i;
typedef __attribute__((ext_vector_type(4))) int v4i;

#define MOE_N 8192
#define MOE_D 1024
#define MOE_E 8
#define MOE_F 1024
#define MOE_CAP 1024
#define KC 64           // k-chunk staged in LDS (64 dwords per row)
#define APAD 4          // LDS row padding (dwords); TDM pad: 4 DW every 64 DW

// ---------------------------------------------------------------------------
// Kernel 1: router.  One wave per token; 8 dot-products of length D, then
// wave32 butterfly reduction and argmax (strict > so first max wins, matching
// jnp.argmax; softmax is monotone so it is skipped).
// ---------------------------------------------------------------------------
__global__ __launch_bounds__(256) void moe_router(
    const float* __restrict__ x, const float* __restrict__ Wr,
    const float* __restrict__ br, int* __restrict__ ids) {
  int token = (blockIdx.x * blockDim.x + threadIdx.x) >> 5;
  int lane = threadIdx.x & 31;
  if (token >= MOE_N) return;
  const float* xr = x + (size_t)token * MOE_D;
  float acc[MOE_E];
#pragma unroll
  for (int e = 0; e < MOE_E; ++e) acc[e] = 0.f;
  for (int k = lane; k < MOE_D; k += 32) {
    float xv = xr[k];
#pragma unroll
    for (int e = 0; e < MOE_E; ++e) acc[e] += xv * Wr[e * MOE_D + k];
  }
#pragma unroll
  for (int e = 0; e < MOE_E; ++e) {
    float v = acc[e];
#pragma unroll
    for (int off = 16; off > 0; off >>= 1) v += __shfl_xor(v, off, 32);
    acc[e] = v;
  }
  if (lane == 0) {
    int best = 0;
    float bv = acc[0] + br[0];
#pragma unroll
    for (int e = 1; e < MOE_E; ++e) {
      float t = acc[e] + br[e];
      if (t > bv) { bv = t; best = e; }
    }
    ids[token] = best;
  }
}

// ---------------------------------------------------------------------------
// Kernel 2: deterministic capacity scan.  One block (1024 thr = 32 waves) per
// expert.  For each 1024-token chunk: wave32 ballot prefix + LDS scan of wave
// totals gives the exact flat-token-order rank.  rank < capacity -> token's
// index appended to perm[e][rank].
// ---------------------------------------------------------------------------
__global__ __launch_bounds__(1024) void moe_scan(
    const int* __restrict__ ids, int* __restrict__ perm, int* __restrict__ cnt) {
  int e = blockIdx.x;
  int tid = threadIdx.x;
  int lane = tid & 31;
  int wid = tid >> 5;  // 0..31
  __shared__ int wsum[32];
  __shared__ int srun;
  if (tid == 0) srun = 0;
  __syncthreads();
  for (int base = 0; base < MOE_N; base += 1024) {
    int n = base + tid;
    int match = (ids[n] == e) ? 1 : 0;
    unsigned mask = (unsigned)__ballot(match);  // wave32: low 32 bits
    int prefix = __popc(mask & ((1u << lane) - 1u));
    if (lane == 0) wsum[wid] = __popc(mask);
    __syncthreads();
    int woff = 0;
    for (int i = 0; i < wid; ++i) woff += wsum[i];
    int rank0 = srun + woff + prefix;  // 0-based rank in flat token order
    if (match && rank0 < MOE_CAP) perm[e * MOE_CAP + rank0] = n;
    __syncthreads();
    if (tid == 0) {
      int tot = 0;
      for (int i = 0; i < 32; ++i) tot += wsum[i];
      srun += tot;
    }
    __syncthreads();
  }
  if (tid == 0) cnt[e] = (srun < MOE_CAP) ? srun : MOE_CAP;
}

// ---------------------------------------------------------------------------
// Kernel 3: passthrough, out = x (dropped tokens keep their input; kept rows
// are overwritten by the GEMM kernel which runs after this on the stream).
// ---------------------------------------------------------------------------
__global__ __launch_bounds__(256) void moe_copy(
    const float4* __restrict__ in, float4* __restrict__ out, int n4) {
  int i = blockIdx.x * blockDim.x + threadIdx.x;
  if (i < n4) out[i] = in[i];
}

// ---------------------------------------------------------------------------
// TDM issue: DMA a 128-row x KC-dword tile of We[e] (row stride D dwords)
// into LDS at `ldsb`, with hardware padding of APAD dwords after every KC
// dwords (so the LDS image matches float[128][KC+APAD]).
// Descriptor per CDNA5 ISA ch.8; issued via inline asm (portable across
// toolchains).  Wave-level op: EXEC ignored, tracked by TENSORcnt.
// ---------------------------------------------------------------------------
__device__ __forceinline__ void tdm_load_b_tile(unsigned ldsb, const float* gptr) {
  unsigned long long ga = (unsigned long long)(size_t)gptr;
  v4u g0;
  g0.x = 1u;                                        // count=1 (valid user D#)
  g0.y = ldsb;                                      // lds_addr (bytes)
  g0.z = (unsigned)(ga & 0xffffffffu);              // global_addr[31:0]
  g0.w = (unsigned)((ga >> 32) & 0x1ffffffu)        // global_addr[56:32]
         | (2u << 30);                              // type=2 ("image")
  v8i g1;
  g1[0] = (int)((2u << 16)                          // data_size = 4 bytes
                | (1u << 20)                        // pad_enable
                | (5u << 22)                        // pad_interval: 64 DW
                | (3u << 25));                      // pad_amount: 4 DW
  g1[1] = (int)((unsigned)MOE_D << 16);             // tensor_dim0[15:0] @ [31:16]
  g1[2] = (int)(((unsigned)MOE_D >> 16)             // tensor_dim0[31:16]
                | ((unsigned)MOE_F << 16));         // tensor_dim1[15:0]
  g1[3] = (int)(((unsigned)MOE_F >> 16)             // tensor_dim1[31:16]
                | ((unsigned)KC << 16));            // tile_dim0 = KC
  g1[4] = 128;                                      // tile_dim1 = 128 rows
  g1[5] = MOE_D;                                    // tensor_dim0_stride[31:0]
  g1[6] = 0;                                        // stride hi / dim1_stride
  g1[7] = 0;
  v4i g2 = {0, 0, 0, 0};
  v4i g3 = {0, 0, 0, 0};
  asm volatile("tensor_load_to_lds %0, %1, %2, %3"
               :
               : "s"(g0), "s"(g1), "s"(g2), "s"(g3)
               : "memory");
}

// ---------------------------------------------------------------------------
// Kernel 4: per-expert GEMM with fp32 WMMA + TDM-staged B, double buffered.
//   grid.x = E * (CAP/16)  -> expert + 16-token row tile
//   grid.y = F / (16*8)    -> 128-column group; 8 waves each own 16 cols
// ---------------------------------------------------------------------------
__global__ __launch_bounds__(256) void moe_gemm(
    const float* __restrict__ x, const float* __restrict__ We,
    const float* __restrict__ be, const int* __restrict__ perm,
    const int* __restrict__ cnt, float* __restrict__ out) {
  const int e = blockIdx.x >> 6;          // / (CAP/16)
  const int mt = blockIdx.x & 63;         // row tile within expert
  const int nrows = cnt[e];
  if (mt * 16 >= nrows) return;
  const int colbase = blockIdx.y * 128;
  const int wid = threadIdx.x >> 5;       // 0..7
  const int lane = threadIdx.x & 31;

  __shared__ float As[16][KC + APAD];
  __shared__ float Bs[2][128][KC + APAD];  // TDM pads rows to KC+APAD dwords
  __shared__ int rows[16];

  if (threadIdx.x < 16) {
    int slot = mt * 16 + threadIdx.x;
    rows[threadIdx.x] = (slot < nrows) ? perm[e * MOE_CAP + slot] : -1;
  }
  __syncthreads();

  v8f c = {};
  const float* Wee = We + (size_t)e * MOE_F * MOE_D;
  const float* Bsrc = Wee + (size_t)colbase * MOE_D;

  // LDS byte offsets of the two B buffers (LDS aperture: low 32 addr bits).
  unsigned bbase0 = (unsigned)(size_t)(__attribute__((address_space(3))) float*)&Bs[0][0][0];
  unsigned bbase1 = (unsigned)(size_t)(__attribute__((address_space(3))) float*)&Bs[1][0][0];

  // Kick off TDM for chunk 0 into buffer 0 (one wave issues the DMA).
  if (wid == 0) tdm_load_b_tile(bbase0, Bsrc);

  int buf = 0;
  for (int k0 = 0; k0 < MOE_D; k0 += KC) {
    // Stage A: 16 gathered token rows, KC contiguous k each (small: 4KB).
    for (int i = threadIdx.x; i < 16 * KC; i += 256) {
      int r = i / KC, kk = i % KC;
      int tok = rows[r];
      As[r][kk] = (tok >= 0) ? x[(size_t)tok * MOE_D + k0 + kk] : 0.f;
    }
    // Prefetch next A chunk (global_prefetch_b8).
    if (k0 + KC < MOE_D && threadIdx.x < 16) {
      int tok = rows[threadIdx.x];
      if (tok >= 0) __builtin_prefetch(&x[(size_t)tok * MOE_D + k0 + KC], 0, 0);
    }
    // Wave 0: wait for this iteration's B tile DMA to land in LDS.
    if (wid == 0) __builtin_amdgcn_s_wait_tensorcnt(0);
    __syncthreads();  // publishes A stores + TDM-written B to all waves

    // Issue TDM for the next chunk into the other buffer (overlaps compute).
    if (wid == 0 && k0 + KC < MOE_D)
      tdm_load_b_tile(buf ? bbase0 : bbase1, Bsrc + k0 + KC);

    const int mycol = wid * 16;
#pragma unroll
    for (int ks = 0; ks < KC; ks += 4) {
      // A 16x4 f32 fragment: lanes 0-15 hold M=lane, K=ks..ks+1;
      // lanes 16-31 hold M=lane-16, K=ks+2..ks+3 (ISA 7.12.2 layout).
      int am = lane & 15;
      int ak = ks + ((lane >> 4) << 1);
      v2f a;
      a.x = As[am][ak];
      a.y = As[am][ak + 1];
      // B 4x16 fragment: row-striped across lanes within a VGPR.
      int bn = lane & 15;
      int bk = ks + (lane >> 4);
      v2f b;
      b.x = Bs[buf][mycol + bn][bk];
      b.y = Bs[buf][mycol + bn][bk + 2];
      c = __builtin_amdgcn_wmma_f32_16x16x4_f32(
          /*neg_a=*/false, a, /*neg_b=*/false, b,
          /*c_mod=*/(short)0, c, /*reuse_a=*/false, /*reuse_b=*/false);
    }
    __syncthreads();  // all waves done reading before As is restaged
    buf ^= 1;
  }

  // Store C (16x16 f32 layout): VGPR j -> M=j (lanes 0-15), M=j+8 (lanes 16-31).
  const int cn = lane & 15;
  const int moff = (lane >> 4) * 8;
#pragma unroll
  for (int j = 0; j < 8; ++j) {
    int r = moff + j;
    int tok = rows[r];
    if (tok >= 0) {
      int fcol = colbase + wid * 16 + cn;
      out[(size_t)tok * MOE_F + fcol] = c[j] + be[e * MOE_F + fcol];
    }
  }
}

// ---------------------------------------------------------------------------
// Host launcher.
// ---------------------------------------------------------------------------
extern "C" void kernel_launch(void* const* d_in, const int* in_sizes, int n_in,
                              void* d_out, int out_size, void* d_ws, size_t ws_size,
                              hipStream_t stream) {
  const float* x = (const float*)d_in[0];   // [N, D]
  const float* Wr = (const float*)d_in[1];  // [E, D]
  const float* br = (const float*)d_in[2];  // [E]
  const float* We = (const float*)d_in[3];  // [E, F, D]
  const float* be = (const float*)d_in[4];  // [E, F]
  // d_in[5] = capacity (scalar, known = 1024)
  float* out = (float*)d_out;

  // Workspace layout: ids[N] | perm[E*CAP] | cnt[E]
  int* ids = (int*)d_ws;
  int* perm = ids + MOE_N;
  int* cnt = perm + MOE_E * MOE_CAP;

  // 1) Router: 8 tokens per 256-thread block (one wave each).
  moe_router<<<MOE_N / 8, 256, 0, stream>>>(x, Wr, br, ids);

  // 2) Deterministic capacity scan: one block per expert.
  moe_scan<<<MOE_E, 1024, 0, stream>>>(ids, perm, cnt);

  // 3) Passthrough copy out = x.
  int n4 = MOE_N * MOE_D / 4;
  moe_copy<<<(n4 + 255) / 256, 256, 0, stream>>>(
      (const float4*)x, (float4*)out, n4);

  // 4) Expert GEMMs: grid (E * CAP/16, F/128), 8 waves per block.
  dim3 ggrid(MOE_E * (MOE_CAP / 16), MOE_F / 128);
  moe_gemm<<<ggrid, 256, 0, stream>>>(x, We, be, perm, cnt, out);
}